// DVQuantumLayer_18863496364266
// MI455X (gfx1250) — compile-verified
//
#include <hip/hip_runtime.h>

#define NQ 12
#define NL 2
#define NSTATE 4096       // 2^12 amplitudes
#define NTHREADS 256      // 8 wave32 waves
#define NGATES (NQ + NL * 2 * NQ)   // 60 gates that need cos/sin(theta/2)

typedef __attribute__((ext_vector_type(2))) float v2f;
typedef __attribute__((ext_vector_type(8))) float v8f;

// ---- RX butterfly on target bit-position tb (2048 pairs, 8/thread) -----
__device__ __forceinline__ void rx_pass(v2f* amp, int tb, float c, float s,
                                        int tid) {
  const int st = 1 << tb;
#pragma unroll
  for (int j = 0; j < NSTATE / (2 * NTHREADS); ++j) {
    int k  = tid + NTHREADS * j;                       // 0..2047
    int i0 = ((k >> tb) << (tb + 1)) | (k & (st - 1)); // insert 0 at tb
    int i1 = i0 + st;
    v2f a0 = amp[i0], a1 = amp[i1];
    v2f n0, n1;
    // RX: a0' = c*a0 - i s*a1 ; a1' = -i s*a0 + c*a1
    n0.x = c * a0.x + s * a1.y;
    n0.y = c * a0.y - s * a1.x;
    n1.x = c * a1.x + s * a0.y;
    n1.y = c * a1.y - s * a0.x;
    amp[i0] = n0;
    amp[i1] = n1;
  }
}

// ---- CRX: control bit cb, target bit tb (1024 pairs, 4/thread) ---------
__device__ __forceinline__ void crx_pass(v2f* amp, int cb, int tb, float c,
                                         float s, int tid) {
  const int p1 = (cb < tb) ? cb : tb;
  const int p2 = (cb < tb) ? tb : cb;
#pragma unroll
  for (int j = 0; j < NSTATE / (4 * NTHREADS); ++j) {
    int k = tid + NTHREADS * j;                        // 0..1023
    int i = ((k >> p1) << (p1 + 1)) | (k & ((1 << p1) - 1));
    i     = ((i >> p2) << (p2 + 1)) | (i & ((1 << p2) - 1));
    int i0 = i | (1 << cb);        // control=1, target=0
    int i1 = i0 | (1 << tb);       // control=1, target=1
    v2f a0 = amp[i0], a1 = amp[i1];
    v2f n0, n1;
    n0.x = c * a0.x + s * a1.y;
    n0.y = c * a0.y - s * a1.x;
    n1.x = c * a1.x + s * a0.y;
    n1.y = c * a1.y - s * a0.x;
    amp[i0] = n0;
    amp[i1] = n1;
  }
}

__global__ __launch_bounds__(NTHREADS) void
qsim12_kernel(const float* __restrict__ x, const float* __restrict__ params,
              float* __restrict__ out) {
  __shared__ float lds[2 * NSTATE];  // amps (v2f[4096]); later probs + scratch
  __shared__ v2f   csT[NGATES];      // (cos, sin)(theta/2) per gate
  __shared__ float phiT[128];        // phihi[64] | philo[64] (per layer)
  v2f* amp = (v2f*)lds;
  const int b   = blockIdx.x;
  const int tid = threadIdx.x;

  // gate-angle table: 12 embedding RX (per-sample) + per layer 12 RX + 12 CRX
  if (tid < NGATES) {
    float th;
    if (tid < NQ) {
      th = x[b * NQ + tid];
    } else {
      int r = tid - NQ;
      int l = r / (2 * NQ);
      int g = r % (2 * NQ);            // 0..11 RX, 12..23 CRX (reuses p[0..11])
      th = params[l * 3 * NQ + (g % NQ)];
    }
    th *= 0.5f;
    v2f t = {cosf(th), sinf(th)};
    csT[tid] = t;
  }
  // |0...0>
  for (int i = tid; i < NSTATE; i += NTHREADS) {
    v2f z = {0.0f, 0.0f};
    amp[i] = z;
  }
  __syncthreads();
  if (tid == 0) {
    v2f one = {1.0f, 0.0f};
    amp[0] = one;
  }
  __syncthreads();

  // AngleEmbedding: RX(x[b,q]) on wire q (bit 11-q)
#pragma unroll
  for (int q = 0; q < NQ; ++q) {
    v2f a = csT[q];
    rx_pass(amp, 11 - q, a.x, a.y, tid);
    __syncthreads();
  }

#pragma unroll
  for (int l = 0; l < NL; ++l) {
    const float* p = params + l * 3 * NQ;

    // RX layer
#pragma unroll
    for (int q = 0; q < NQ; ++q) {
      v2f a = csT[NQ + l * 2 * NQ + q];
      rx_pass(amp, 11 - q, a.x, a.y, tid);
      __syncthreads();
    }

    // fused RZ layer: amp *= exp(-i*phi), phi = phihi[hi6] + philo[lo6]
    if (tid < 128) {
      int idx = tid & 63;
      float acc = 0.0f;
      if (tid < 64) {                      // phihi: q=0..5 -> bit (5-q) of hi
#pragma unroll
        for (int q = 0; q < 6; ++q) {
          float h = 0.5f * p[NQ + q];
          acc += ((idx >> (5 - q)) & 1) ? -h : h;
        }
      } else {                             // philo: q=6..11 -> bit (11-q) of lo
#pragma unroll
        for (int q = 6; q < 12; ++q) {
          float h = 0.5f * p[NQ + q];
          acc += ((idx >> (11 - q)) & 1) ? -h : h;
        }
      }
      phiT[tid] = acc;
    }
    __syncthreads();
#pragma unroll
    for (int j = 0; j < NSTATE / NTHREADS; ++j) {
      int i = tid + NTHREADS * j;
      float phi = phiT[i >> 6] + phiT[64 + (i & 63)];
      float cp = cosf(phi), sp = sinf(phi);
      v2f a = amp[i];
      v2f n;
      n.x = a.x * cp + a.y * sp;
      n.y = a.y * cp - a.x * sp;
      amp[i] = n;
    }
    __syncthreads();

    // CRX ring (param_counter reset: reuses p[0..11])
    {
      v2f a = csT[NQ + l * 2 * NQ + NQ + 0];
      crx_pass(amp, /*cb: ctrl wire 11*/ 0, /*tb: tgt wire 0*/ 11, a.x, a.y,
               tid);
      __syncthreads();
    }
#pragma unroll
    for (int g = 1; g < NQ; ++g) {         // ctrl wire 11-g -> cb=g ; tgt wire 12-g -> tb=g-1
      v2f a = csT[NQ + l * 2 * NQ + NQ + g];
      crx_pass(amp, g, g - 1, a.x, a.y, tid);
      __syncthreads();
    }
  }

  // probabilities: stage in registers, then repurpose lds[0..4095]
  float pr[NSTATE / NTHREADS];
#pragma unroll
  for (int j = 0; j < NSTATE / NTHREADS; ++j) {
    v2f a = amp[tid + NTHREADS * j];
    pr[j] = a.x * a.x + a.y * a.y;
  }
  __syncthreads();
  float* prob = lds;                       // 16 KB, overlays dead amp storage
  float* red  = lds + NSTATE;              // scratch in dead second half
#pragma unroll
  for (int j = 0; j < NSTATE / NTHREADS; ++j)
    prob[tid + NTHREADS * j] = pr[j];
  __syncthreads();

  // ---- readout: out(q) = sum_hi SH(hi,q) * [P(64x64) x SL(64x16)](hi,q)
  // Inner GEMM via V_WMMA_F32_16X16X4_F32; waves 0..3 take M-tiles 0..3.
  // Branchless signs: bit(11-q) of lo == bit 11 of (lo<<q); bit(5-q) of hi
  // == bit 5 of (hi<<q); out-of-range bit indices yield 0 -> sign +1.
  const int wave = tid >> 5;
  const int lane = tid & 31;
  if (wave < 4) {
    const int m   = lane & 15;             // M row; also N column q for B synth
    const int kb  = (lane >> 4) << 1;      // lanes 0-15: K 0,1 ; 16-31: K 2,3
    const int row = (wave << 4) + m;       // hi index
    v8f acc = {0.f, 0.f, 0.f, 0.f, 0.f, 0.f, 0.f, 0.f};
#pragma unroll
    for (int kc = 0; kc < 16; ++kc) {
      int col = (kc << 2) + kb;            // lo index
      v2f a = *(const v2f*)&prob[(row << 6) + col];
      v2f bm;
      bm.x = 1.0f - 2.0f * (float)(((col << m) >> 11) & 1);
      bm.y = 1.0f - 2.0f * (float)((((col + 1) << m) >> 11) & 1);
      acc = __builtin_amdgcn_wmma_f32_16x16x4_f32(
          false, a, false, bm, (short)0, acc, false, false);
    }
    // D layout: lane l, vgpr r -> M = r + 8*(l>=16), N = l&15.
    float partial = 0.0f;
#pragma unroll
    for (int r = 0; r < 8; ++r) {
      int hi = (wave << 4) + r + ((lane >> 4) << 3);
      partial += acc[r] * (1.0f - 2.0f * (float)(((hi << m) >> 5) & 1));
    }
    red[(wave << 5) + lane] = partial;
  }
  __syncthreads();

  if (tid < NQ) {
    float o = 0.0f;
#pragma unroll
    for (int w = 0; w < 4; ++w)
      o += red[(w << 5) + tid] + red[(w << 5) + tid + 16];
    out[b * NQ + tid] = o;
  }
}

extern "C" void kernel_launch(void* const* d_in, const int* in_sizes, int n_in,
                              void* d_out, int out_size, void* d_ws,
                              size_t ws_size, hipStream_t stream) {
  const float* x      = (const float*)d_in[0];  // (B, 12) f32
  const float* params = (const float*)d_in[1];  // (2, 36) f32
  float* out          = (float*)d_out;          // (B, 12) f32
  const int Bn = in_sizes[0] / NQ;
  qsim12_kernel<<<dim3(Bn), dim3(NTHREADS), 0, stream>>>(x, params, out);
}